// Encoder_woEdgeConv_46420006535585
// MI455X (gfx1250) — compile-verified
//
#include <hip/hip_runtime.h>

typedef __attribute__((ext_vector_type(2))) float v2f;
typedef __attribute__((ext_vector_type(8))) float v8f;

static constexpr int   Bn   = 4;
static constexpr int   Nn   = 16384;
static constexpr float EPSBN = 1e-5f;

// ---------------------------------------------------------------------------
// Generic fused GEMM:  y[b, co, n] = sum_ci W[co,ci] * f(in[b,ci,n]) + bias[co]
// where f applies the previous layer's batchnorm scale/shift (+ optional ReLU).
// Optionally accumulates per-output-channel sum / sumsq for this layer's BN.
// One wave computes a 16(M) x 16(cols) fp32 tile via V_WMMA_F32_16X16X4_F32.
// ---------------------------------------------------------------------------
template<int CIN, bool APPLY, bool RELU>
__device__ __forceinline__ float load_in(const float* __restrict__ in,
                                         const float* __restrict__ sc,
                                         const float* __restrict__ sh,
                                         int c, int n) {
  if (c >= CIN) return 0.0f;
  float v = in[(long)c * Nn + n];
  if constexpr (APPLY) v = sc[c] * v + sh[c];
  if constexpr (RELU)  v = fmaxf(v, 0.0f);
  return v;
}

template<int CIN, int COUT, bool APPLY, bool RELU, bool STATS>
__global__ void __launch_bounds__(256)
gemm_bn_kernel(const float* __restrict__ in, long inBatchStride,
               const float* __restrict__ W, const float* __restrict__ bias,
               const float* __restrict__ sc, const float* __restrict__ sh,
               float* __restrict__ outp, long outBatchStride,
               float* __restrict__ stats)
{
  constexpr int CINP = (CIN + 3) & ~3;
  __shared__ float ssum[16];
  __shared__ float ssq[16];
  if constexpr (STATS) {
    if (threadIdx.x < 16) { ssum[threadIdx.x] = 0.0f; ssq[threadIdx.x] = 0.0f; }
    __syncthreads();
  }
  const int lane = threadIdx.x & 31;
  const int wave = threadIdx.x >> 5;
  const int l    = lane & 15;
  const int half = lane >> 4;
  const int n0   = blockIdx.x * 128 + wave * 16;   // 8 waves x 16 cols
  const int m0   = blockIdx.y * 16;                // 16 out-channels
  const int b    = blockIdx.z;
  const float* inB = in + (long)b * inBatchStride;

  v8f acc = {0.f,0.f,0.f,0.f,0.f,0.f,0.f,0.f};
  for (int k0 = 0; k0 < CINP; k0 += 4) {
    const int ka = k0 + 2 * half;
    const int kb = ka + 1;
    v2f av, bv;
    // A (16x4 f32): lanes 0-15 -> K=0,1 ; lanes 16-31 -> K=2,3 ; M = lane&15
    av.x = (ka < CIN) ? W[(m0 + l) * CIN + ka] : 0.0f;
    av.y = (kb < CIN) ? W[(m0 + l) * CIN + kb] : 0.0f;
    // B (4x16 f32): row-striped across lanes; halves hold K and K+2
    bv.x = load_in<CIN, APPLY, RELU>(inB, sc, sh, ka, n0 + l);
    bv.y = load_in<CIN, APPLY, RELU>(inB, sc, sh, kb, n0 + l);
    acc = __builtin_amdgcn_wmma_f32_16x16x4_f32(false, av, false, bv,
                                                (short)0, acc, false, false);
  }
  float* outB = outp + (long)b * outBatchStride;
#pragma unroll
  for (int r = 0; r < 8; ++r) {
    const int c = m0 + r + 8 * half;   // C/D layout: VGPR r -> M = r (+8 for hi half)
    const float val = acc[r] + bias[c];
    outB[(long)c * Nn + n0 + l] = val;
    if constexpr (STATS) {
      atomicAdd(&ssum[r + 8 * half], val);
      atomicAdd(&ssq [r + 8 * half], val * val);
    }
  }
  if constexpr (STATS) {
    __syncthreads();
    if (threadIdx.x < 16) {
      atomicAdd(&stats[m0 + threadIdx.x],        ssum[threadIdx.x]);
      atomicAdd(&stats[COUT + m0 + threadIdx.x], ssq[threadIdx.x]);
    }
  }
}

// ---------------------------------------------------------------------------
// BN finalize: per-channel scale/shift from accumulated sum/sumsq.
// ---------------------------------------------------------------------------
__global__ void bn_finalize_kernel(const float* __restrict__ stats,
                                   const float* __restrict__ gamma,
                                   const float* __restrict__ beta,
                                   float* __restrict__ sc, float* __restrict__ sh,
                                   int C, float inv)
{
  int c = blockIdx.x * blockDim.x + threadIdx.x;
  if (c >= C) return;
  float m  = stats[c] * inv;
  float v  = stats[C + c] * inv - m * m;
  float rs = rsqrtf(v + EPSBN);
  float s  = gamma[c] * rs;
  sc[c] = s;
  sh[c] = beta[c] - m * s;
}

// ---------------------------------------------------------------------------
// point_feat = relu(BN2(y2)) + BNr(yr)  -> written to out channels [0,128)
// ---------------------------------------------------------------------------
__global__ void pointfeat_kernel(const float* __restrict__ y2,
                                 const float* __restrict__ yr,
                                 const float* __restrict__ s2, const float* __restrict__ t2,
                                 const float* __restrict__ sr, const float* __restrict__ tr,
                                 float* __restrict__ out)
{
  long i = (long)blockIdx.x * blockDim.x + threadIdx.x;
  const long total = (long)Bn * 128 * Nn;
  if (i >= total) return;
  int n = (int)(i % Nn);
  int c = (int)((i / Nn) % 128);
  int b = (int)(i / (128L * Nn));
  float h = fmaxf(s2[c] * y2[i] + t2[c], 0.0f);
  float p = sr[c] * yr[i] + tr[c];
  out[(long)b * 512 * Nn + (long)c * Nn + n] = h + p;
}

// ---------------------------------------------------------------------------
// local_feat[b,c,n] = max_{k<16} pf[b,c,idx[b,n,k]] -> out channels [128,256)
// ---------------------------------------------------------------------------
__global__ void knnmax_kernel(const long long* __restrict__ idx,
                              float* __restrict__ out)
{
  long i = (long)blockIdx.x * blockDim.x + threadIdx.x;
  const long total = (long)Bn * 128 * Nn;
  if (i >= total) return;
  int n = (int)(i % Nn);
  int c = (int)((i / Nn) % 128);
  int b = (int)(i / (128L * Nn));
  const float* pf = out + (long)b * 512 * Nn + (long)c * Nn;
  const long long* ip = idx + ((long)b * Nn + n) * 20;
  float m = pf[(int)ip[0]];
#pragma unroll
  for (int k = 1; k < 16; ++k) m = fmaxf(m, pf[(int)ip[k]]);
  out[(long)b * 512 * Nn + (long)(128 + c) * Nn + n] = m;
}

// ---------------------------------------------------------------------------
// Softmax over the 64 cluster logits per column; also accumulates
// asum[b,k] = sum_n a[b,k,n] via LDS + global float atomics.
// ---------------------------------------------------------------------------
__global__ void softmax_assign_kernel(float* __restrict__ logits,
                                      float* __restrict__ asum)
{
  __shared__ float ls[64];
  long col = (long)blockIdx.x * 256 + threadIdx.x;   // 256 | Nn, same b per block
  int b = (int)(col / Nn);
  int n = (int)(col % Nn);
  if (threadIdx.x < 64) ls[threadIdx.x] = 0.0f;
  __syncthreads();
  float* base = logits + (long)b * 64 * Nn + n;
  float mx = base[0];
  for (int k = 1; k < 64; ++k) mx = fmaxf(mx, base[(long)k * Nn]);
  float s = 0.0f;
  for (int k = 0; k < 64; ++k) s += __expf(base[(long)k * Nn] - mx);
  float inv = 1.0f / s;
  for (int k = 0; k < 64; ++k) {
    float a = __expf(base[(long)k * Nn] - mx) * inv;
    base[(long)k * Nn] = a;
    atomicAdd(&ls[k], a);
  }
  __syncthreads();
  if (threadIdx.x < 64) atomicAdd(&asum[b * 64 + threadIdx.x], ls[threadIdx.x]);
}

// ---------------------------------------------------------------------------
// VLAD aggregation: vlad[b,k,d] += sum_n a[b,k,n]*feat[b,d,n].
// M=64 (clusters), Ncols=256 (dims), K=16384 split into 8 chunks (atomic acc).
// ---------------------------------------------------------------------------
__global__ void __launch_bounds__(256)
vlad_gemm_kernel(const float* __restrict__ a, const float* __restrict__ out,
                 float* __restrict__ vlad)
{
  const int lane = threadIdx.x & 31;
  const int wave = threadIdx.x >> 5;
  const int l    = lane & 15;
  const int half = lane >> 4;
  const int tile = blockIdx.x * 8 + wave;   // 0..63
  const int k0   = (tile >> 4) * 16;        // 4 cluster tiles
  const int d0   = (tile & 15) * 16;        // 16 dim tiles
  const int b    = blockIdx.z;
  const int nStart = blockIdx.y * (Nn / 8);
  const float* aB = a   + (long)b * 64  * Nn;
  const float* fB = out + (long)b * 512 * Nn;   // feat = channels 0..255

  v8f acc = {0.f,0.f,0.f,0.f,0.f,0.f,0.f,0.f};
  for (int n = nStart; n < nStart + Nn / 8; n += 4) {
    const int na = n + 2 * half;
    v2f av, bv;
    av.x = aB[(long)(k0 + l) * Nn + na];
    av.y = aB[(long)(k0 + l) * Nn + na + 1];
    bv.x = fB[(long)(d0 + l) * Nn + na];
    bv.y = fB[(long)(d0 + l) * Nn + na + 1];
    acc = __builtin_amdgcn_wmma_f32_16x16x4_f32(false, av, false, bv,
                                                (short)0, acc, false, false);
  }
#pragma unroll
  for (int r = 0; r < 8; ++r)
    atomicAdd(&vlad[((long)b * 64 + k0 + r + 8 * half) * 256 + d0 + l], acc[r]);
}

// ---------------------------------------------------------------------------
// Per-(b,k) row: subtract asum*centroid, L2-normalize -> v.
// ---------------------------------------------------------------------------
__global__ void vlad_norm_kernel(const float* __restrict__ vlad,
                                 const float* __restrict__ asum,
                                 const float* __restrict__ cent,
                                 float* __restrict__ v)
{
  __shared__ float red[256];
  const int k = blockIdx.x, b = blockIdx.y, d = threadIdx.x;
  float val = vlad[((long)b * 64 + k) * 256 + d] - asum[b * 64 + k] * cent[k * 256 + d];
  red[d] = val * val;
  __syncthreads();
  for (int s = 128; s > 0; s >>= 1) {
    if (d < s) red[d] += red[d + s];
    __syncthreads();
  }
  float norm = sqrtf(red[0]) + 1e-12f;
  v[(long)b * 16384 + k * 256 + d] = val / norm;
}

__global__ void vnorm_kernel(const float* __restrict__ v, float* __restrict__ vnorm)
{
  __shared__ float red[256];
  const int b = blockIdx.x;
  float s = 0.0f;
  for (int i = threadIdx.x; i < 16384; i += 256) {
    float x = v[(long)b * 16384 + i];
    s += x * x;
  }
  red[threadIdx.x] = s;
  __syncthreads();
  for (int st = 128; st > 0; st >>= 1) {
    if (threadIdx.x < st) red[threadIdx.x] += red[threadIdx.x + st];
    __syncthreads();
  }
  if (threadIdx.x == 0) vnorm[b] = sqrtf(red[0]) + 1e-12f;
}

// g_raw[b,o] = dot(Wp[o,:], v[b,:]) / vnorm[b] + bp[o]
__global__ void gdot_kernel(const float* __restrict__ Wp, const float* __restrict__ v,
                            const float* __restrict__ vnorm, const float* __restrict__ bp,
                            float* __restrict__ gacc)
{
  __shared__ float red[256];
  const int o = blockIdx.x, b = blockIdx.y;
  const float* wr = Wp + (long)o * 16384;
  const float* vb = v  + (long)b * 16384;
  float s = 0.0f;
  for (int i = threadIdx.x; i < 16384; i += 256) s += wr[i] * vb[i];
  red[threadIdx.x] = s;
  __syncthreads();
  for (int st = 128; st > 0; st >>= 1) {
    if (threadIdx.x < st) red[threadIdx.x] += red[threadIdx.x + st];
    __syncthreads();
  }
  if (threadIdx.x == 0) gacc[b * 256 + o] = red[0] / vnorm[b] + bp[o];
}

// BN of g over the batch dim (4 samples per channel).
__global__ void gbn_kernel(const float* __restrict__ gacc,
                           const float* __restrict__ gp, const float* __restrict__ bep,
                           float* __restrict__ gfinal)
{
  const int o = threadIdx.x;
  float m = 0.0f, q = 0.0f;
  for (int b = 0; b < 4; ++b) { float x = gacc[b * 256 + o]; m += x; q += x * x; }
  m *= 0.25f;
  q = q * 0.25f - m * m;
  float rs = rsqrtf(q + EPSBN);
  for (int b = 0; b < 4; ++b)
    gfinal[b * 256 + o] = gp[o] * (gacc[b * 256 + o] - m) * rs + bep[o];
}

__global__ void bcast_kernel(const float* __restrict__ gfinal, float* __restrict__ out)
{
  long i = (long)blockIdx.x * blockDim.x + threadIdx.x;
  const long total = (long)Bn * 256 * Nn;
  if (i >= total) return;
  int n = (int)(i % Nn);
  int o = (int)((i / Nn) % 256);
  int b = (int)(i / (256L * Nn));
  out[(long)b * 512 * Nn + (long)(256 + o) * Nn + n] = gfinal[b * 256 + o];
}

// ---------------------------------------------------------------------------
extern "C" void kernel_launch(void* const* d_in, const int* in_sizes, int n_in,
                              void* d_out, int out_size, void* d_ws, size_t ws_size,
                              hipStream_t stream)
{
  const float*     x    = (const float*)d_in[0];
  const long long* knn  = (const long long*)d_in[1];
  const float *W0 = (const float*)d_in[2],  *b0 = (const float*)d_in[3];
  const float *g0 = (const float*)d_in[4],  *be0= (const float*)d_in[5];
  const float *W1 = (const float*)d_in[6],  *b1 = (const float*)d_in[7];
  const float *g1 = (const float*)d_in[8],  *be1= (const float*)d_in[9];
  const float *W2 = (const float*)d_in[10], *b2 = (const float*)d_in[11];
  const float *g2 = (const float*)d_in[12], *be2= (const float*)d_in[13];
  const float *Wr = (const float*)d_in[14], *br = (const float*)d_in[15];
  const float *gr = (const float*)d_in[16], *ber= (const float*)d_in[17];
  const float *Wa = (const float*)d_in[18], *ba = (const float*)d_in[19];
  const float *cent = (const float*)d_in[20];
  const float *Wp = (const float*)d_in[21], *bp = (const float*)d_in[22];
  const float *gp = (const float*)d_in[23], *bep= (const float*)d_in[24];
  float* out = (float*)d_out;

  // workspace layout (floats)
  float* w      = (float*)d_ws;
  float* y0     = w;                    // 4*64*N   = 4,194,304
  float* y1     = y0 + 4194304;         // 4*128*N  = 8,388,608
  float* yr     = y1 + 8388608;
  float* y2     = yr + 8388608;
  float* zbase  = y2 + 8388608;         // zero-initialized accumulator region:
  float* stats0 = zbase;                //   128
  float* stats1 = stats0 + 128;         //   256
  float* statsr = stats1 + 256;         //   256
  float* stats2 = statsr + 256;         //   256
  float* asum   = stats2 + 256;         //   256
  float* vlad   = asum + 256;           //   65536
  float* s0     = vlad + 65536;         // end of zero region (66,688 floats)
  float* t0     = s0 + 64;
  float* s1     = t0 + 64;
  float* t1     = s1 + 128;
  float* sr_    = t1 + 128;
  float* tr_    = sr_ + 128;
  float* s2     = tr_ + 128;
  float* t2     = s2 + 128;
  float* vbuf   = t2 + 128;             // 65536
  float* vnorm  = vbuf + 65536;         // 4
  float* gacc   = vnorm + 4;            // 1024
  float* gfinal = gacc + 1024;          // 1024
  float* logits = y1;                   // alias: y1 dead after conv2

  hipMemsetAsync(zbase, 0, 66688 * sizeof(float), stream);

  const float invBN = 1.0f / (float)(Bn * Nn);

  // conv0: 6 -> 64 (raw + stats)
  gemm_bn_kernel<6, 64, false, false, true><<<dim3(128, 4, Bn), 256, 0, stream>>>(
      x, 6L * Nn, W0, b0, nullptr, nullptr, y0, 64L * Nn, stats0);
  bn_finalize_kernel<<<1, 64, 0, stream>>>(stats0, g0, be0, s0, t0, 64, invBN);

  // conv1: 64 -> 128 on relu(BN0(y0))
  gemm_bn_kernel<64, 128, true, true, true><<<dim3(128, 8, Bn), 256, 0, stream>>>(
      y0, 64L * Nn, W1, b1, s0, t0, y1, 128L * Nn, stats1);
  // residual branch: 64 -> 128 on same input
  gemm_bn_kernel<64, 128, true, true, true><<<dim3(128, 8, Bn), 256, 0, stream>>>(
      y0, 64L * Nn, Wr, br, s0, t0, yr, 128L * Nn, statsr);
  bn_finalize_kernel<<<1, 128, 0, stream>>>(stats1, g1, be1, s1, t1, 128, invBN);
  bn_finalize_kernel<<<1, 128, 0, stream>>>(statsr, gr, ber, sr_, tr_, 128, invBN);

  // conv2: 128 -> 128 on relu(BN1(y1))
  gemm_bn_kernel<128, 128, true, true, true><<<dim3(128, 8, Bn), 256, 0, stream>>>(
      y1, 128L * Nn, W2, b2, s1, t1, y2, 128L * Nn, stats2);
  bn_finalize_kernel<<<1, 128, 0, stream>>>(stats2, g2, be2, s2, t2, 128, invBN);

  // point_feat -> out[:, 0:128, :]
  pointfeat_kernel<<<(Bn * 128 * Nn) / 256, 256, 0, stream>>>(
      y2, yr, s2, t2, sr_, tr_, out);
  // KNN max -> out[:, 128:256, :]
  knnmax_kernel<<<(Bn * 128 * Nn) / 256, 256, 0, stream>>>(knn, out);

  // assignment logits: Wa(64x256) * feat
  gemm_bn_kernel<256, 64, false, false, false><<<dim3(128, 4, Bn), 256, 0, stream>>>(
      out, 512L * Nn, Wa, ba, nullptr, nullptr, logits, 64L * Nn, nullptr);
  softmax_assign_kernel<<<(Bn * Nn) / 256, 256, 0, stream>>>(logits, asum);

  // vlad accumulation (WMMA, split-K over n with atomic reduce)
  vlad_gemm_kernel<<<dim3(8, 8, Bn), 256, 0, stream>>>(logits, out, vlad);

  vlad_norm_kernel<<<dim3(64, Bn), 256, 0, stream>>>(vlad, asum, cent, vbuf);
  vnorm_kernel<<<Bn, 256, 0, stream>>>(vbuf, vnorm);
  gdot_kernel<<<dim3(256, Bn), 256, 0, stream>>>(Wp, vbuf, vnorm, bp, gacc);
  gbn_kernel<<<1, 256, 0, stream>>>(gacc, gp, bep, gfinal);
  bcast_kernel<<<(Bn * 256 * Nn) / 256, 256, 0, stream>>>(gfinal, out);
}